// PCEncoder_65833258713754
// MI455X (gfx1250) — compile-verified
//
#include <hip/hip_runtime.h>

typedef float v2f __attribute__((ext_vector_type(2)));
typedef float v8f __attribute__((ext_vector_type(8)));

constexpr int Bc  = 8;
constexpr int Nc  = 16384;
constexpr int MAc = 256;
constexpr int MBc = 64;

// ---------------------------------------------------------------------------
// Utility: zero fill
// ---------------------------------------------------------------------------
__global__ void zero_kernel(float* __restrict__ p, unsigned n) {
  unsigned i = blockIdx.x * blockDim.x + threadIdx.x;
  if (i < n) p[i] = 0.0f;
}

// ---------------------------------------------------------------------------
// Stage 1: nearest-3 node_a per point + segment sums (position + count)
// ---------------------------------------------------------------------------
__global__ void nearest3_kernel(const float* __restrict__ pc,
                                const float* __restrict__ node_a,
                                float* __restrict__ min_k_out,   // (B,N,3) as float
                                int*   __restrict__ min_idx,     // (B,N)
                                float* __restrict__ sums) {      // (B,MA,4)
  __shared__ float na[3 * MAc];
  const int b = blockIdx.y;
  for (int t = threadIdx.x; t < 3 * MAc; t += blockDim.x)
    na[t] = node_a[(size_t)b * 3 * MAc + t];
  __syncthreads();
  const int n = blockIdx.x * blockDim.x + threadIdx.x;
  const float px = pc[((size_t)b * 3 + 0) * Nc + n];
  const float py = pc[((size_t)b * 3 + 1) * Nc + n];
  const float pz = pc[((size_t)b * 3 + 2) * Nc + n];
  float d0 = 3.4e38f, d1 = 3.4e38f, d2 = 3.4e38f;
  int i0 = 0, i1 = 0, i2 = 0;
  for (int m = 0; m < MAc; ++m) {
    float dx = px - na[m];
    float dy = py - na[MAc + m];
    float dz = pz - na[2 * MAc + m];
    float d = dx * dx + dy * dy + dz * dz;
    if (d < d0)      { d2 = d1; i2 = i1; d1 = d0; i1 = i0; d0 = d; i0 = m; }
    else if (d < d1) { d2 = d1; i2 = i1; d1 = d;  i1 = m; }
    else if (d < d2) { d2 = d;  i2 = m; }
  }
  const size_t o = ((size_t)b * Nc + n) * 3;
  min_k_out[o + 0] = (float)i0;
  min_k_out[o + 1] = (float)i1;
  min_k_out[o + 2] = (float)i2;
  min_idx[(size_t)b * Nc + n] = i0;
  float* s = sums + ((size_t)b * MAc + i0) * 4;
  atomicAdd(s + 0, px);
  atomicAdd(s + 1, py);
  atomicAdd(s + 2, pz);
  atomicAdd(s + 3, 1.0f);
}

// ---------------------------------------------------------------------------
// Stage 2: cluster means
// ---------------------------------------------------------------------------
__global__ void cluster_mean_kernel(const float* __restrict__ sums,
                                    float* __restrict__ cm) {   // (B,3,MA)
  int t = blockIdx.x * blockDim.x + threadIdx.x;
  if (t < Bc * MAc) {
    int b = t / MAc, m = t % MAc;
    const float* s = sums + (size_t)t * 4;
    float inv = 1.0f / (s[3] + 1e-5f);
    cm[((size_t)b * 3 + 0) * MAc + m] = s[0] * inv;
    cm[((size_t)b * 3 + 1) * MAc + m] = s[1] * inv;
    cm[((size_t)b * 3 + 2) * MAc + m] = s[2] * inv;
  }
}

// ---------------------------------------------------------------------------
// Stage 3: pc_centers + augmented point features (B,8,N)
// ---------------------------------------------------------------------------
__global__ void pcaug_kernel(const float* __restrict__ pc,
                             const float* __restrict__ intensity,
                             const float* __restrict__ sn,
                             const float* __restrict__ label,
                             const float* __restrict__ cm,
                             const int*   __restrict__ min_idx,
                             float* __restrict__ pc_centers,  // (B,3,N)
                             float* __restrict__ aug) {       // (B,8,N)
  int t = blockIdx.x * blockDim.x + threadIdx.x;
  if (t >= Bc * Nc) return;
  int b = t / Nc, n = t % Nc;
  int m = min_idx[t];
  for (int c = 0; c < 3; ++c) {
    float ctr = cm[((size_t)b * 3 + c) * MAc + m];
    pc_centers[((size_t)b * 3 + c) * Nc + n] = ctr;
    aug[((size_t)b * 8 + c) * Nc + n] = pc[((size_t)b * 3 + c) * Nc + n] - ctr;
  }
  aug[((size_t)b * 8 + 3) * Nc + n] = intensity[(size_t)b * Nc + n];
  for (int c = 0; c < 3; ++c)
    aug[((size_t)b * 8 + 4 + c) * Nc + n] = sn[((size_t)b * 3 + c) * Nc + n];
  aug[((size_t)b * 8 + 7) * Nc + n] = label[(size_t)b * Nc + n];
}

// ---------------------------------------------------------------------------
// Generic WMMA fp32 GEMM:  Y[b,o,n] = relu( sum_i W[o,i]*X[b,i,n] + bias[o] )
// Each wave computes a 16(M) x 64(N) tile with 4 accumulators so one A-tile
// feeds 4 back-to-back V_WMMA_F32_16X16X4_F32. Main K loop is unmasked (only
// layers with I%4!=0 take the single masked tail step; branch is wave-uniform
// so EXEC stays all-ones at every WMMA).
// Grid: (Ncols/64, O/16, B), block: 32 (one wave).
// ---------------------------------------------------------------------------
__global__ void wmma_gemm_bias_relu(const float* __restrict__ X,
                                    const float* __restrict__ W,
                                    const float* __restrict__ bias,
                                    float* __restrict__ Y,
                                    int O, int I, int Ncols) {
  const int lane = threadIdx.x;        // 0..31
  const int half = lane >> 4;          // 0 or 1
  const int lm   = lane & 15;
  const int no   = blockIdx.x;
  const int mo   = blockIdx.y;
  const int b    = blockIdx.z;
  const float* Xb = X + (size_t)b * I * Ncols;
  float*       Yb = Y + (size_t)b * O * Ncols;
  const int col0 = no * 64 + lm;
  const int arow = mo * 16 + lm;
  const float* Wr = W + (size_t)arow * I;   // row of weights for this lane

  v8f acc0 = {}, acc1 = {}, acc2 = {}, acc3 = {};
  const int kmain = I >> 2;                 // full (unmasked) K-steps of 4

  for (int kk = 0; kk < kmain; ++kk) {
    // A: 16x4 fp32 -> lane L: VGPR0 = K=half*2, VGPR1 = K=half*2+1 (contiguous)
    const int k0 = kk * 4 + half * 2;
    v2f a;
    a.x = Wr[k0];
    a.y = Wr[k0 + 1];
    // B: 4x16 fp32 -> VGPR r holds rows K=r (lanes 0-15) / K=r+2 (lanes 16-31)
    const float* x0 = Xb + (size_t)k0 * Ncols + col0;
    const float* x1 = x0 + Ncols;
    __builtin_prefetch(x0 + (size_t)16 * Ncols, 0, 1);   // global_prefetch_b8
    v2f b0, b1, b2, b3;
    b0.x = x0[0];  b0.y = x1[0];
    b1.x = x0[16]; b1.y = x1[16];
    b2.x = x0[32]; b2.y = x1[32];
    b3.x = x0[48]; b3.y = x1[48];
    acc0 = __builtin_amdgcn_wmma_f32_16x16x4_f32(false, a, false, b0, (short)0, acc0, false, false);
    acc1 = __builtin_amdgcn_wmma_f32_16x16x4_f32(false, a, false, b1, (short)0, acc1, false, false);
    acc2 = __builtin_amdgcn_wmma_f32_16x16x4_f32(false, a, false, b2, (short)0, acc2, false, false);
    acc3 = __builtin_amdgcn_wmma_f32_16x16x4_f32(false, a, false, b3, (short)0, acc3, false, false);
  }

  if (I & 3) {                              // masked tail step (wave-uniform branch)
    const int k0 = kmain * 4 + half * 2;
    const int k1 = k0 + 1;
    const float m0 = (k0 < I) ? 1.0f : 0.0f;
    const float m1 = (k1 < I) ? 1.0f : 0.0f;
    const int k0c = (k0 < I) ? k0 : (I - 1);
    const int k1c = (k1 < I) ? k1 : (I - 1);
    v2f a;
    a.x = Wr[k0c] * m0;
    a.y = Wr[k1c] * m1;
    const float* x0 = Xb + (size_t)k0c * Ncols + col0;
    const float* x1 = Xb + (size_t)k1c * Ncols + col0;
    v2f b0, b1, b2, b3;
    b0.x = x0[0]  * m0; b0.y = x1[0]  * m1;
    b1.x = x0[16] * m0; b1.y = x1[16] * m1;
    b2.x = x0[32] * m0; b2.y = x1[32] * m1;
    b3.x = x0[48] * m0; b3.y = x1[48] * m1;
    acc0 = __builtin_amdgcn_wmma_f32_16x16x4_f32(false, a, false, b0, (short)0, acc0, false, false);
    acc1 = __builtin_amdgcn_wmma_f32_16x16x4_f32(false, a, false, b1, (short)0, acc1, false, false);
    acc2 = __builtin_amdgcn_wmma_f32_16x16x4_f32(false, a, false, b2, (short)0, acc2, false, false);
    acc3 = __builtin_amdgcn_wmma_f32_16x16x4_f32(false, a, false, b3, (short)0, acc3, false, false);
  }

#pragma unroll
  for (int r = 0; r < 8; ++r) {
    const int row = mo * 16 + half * 8 + r;  // C/D layout: vgpr r -> M=r (+8 hi half)
    const float bs = bias[row];
    float* yr = Yb + (size_t)row * Ncols + col0;
    float v0 = acc0[r] + bs; yr[0]  = v0 > 0.0f ? v0 : 0.0f;
    float v1 = acc1[r] + bs; yr[16] = v1 > 0.0f ? v1 : 0.0f;
    float v2 = acc2[r] + bs; yr[32] = v2 > 0.0f ? v2 : 0.0f;
    float v3 = acc3[r] + bs; yr[48] = v3 > 0.0f ? v3 : 0.0f;
  }
}

// ---------------------------------------------------------------------------
// Segment max over points -> clusters (values are post-ReLU >= 0, so uint
// atomicMax over float bit patterns is order-correct; 0-init == empty-cluster 0)
// ---------------------------------------------------------------------------
__global__ void segmax_kernel(const float* __restrict__ F,   // (B,C,N)
                              const int*   __restrict__ min_idx,
                              unsigned*    __restrict__ M,    // (B,C,MA)
                              int C) {
  int t = blockIdx.x * blockDim.x + threadIdx.x;
  if (t >= Bc * Nc) return;
  int b = t / Nc, n = t % Nc;
  int m = min_idx[t];
  for (int c = 0; c < C; ++c) {
    float v = F[((size_t)b * C + c) * Nc + n];
    atomicMax(&M[((size_t)b * C + c) * MAc + m], __float_as_uint(v));
  }
}

// ---------------------------------------------------------------------------
// Build x2 = concat(f1, m1[min_idx])  (B,64,N)
// ---------------------------------------------------------------------------
__global__ void x2_kernel(const float* __restrict__ f1,   // (B,32,N)
                          const float* __restrict__ m1,   // (B,32,MA)
                          const int*   __restrict__ min_idx,
                          float* __restrict__ x2) {       // (B,64,N)
  int t = blockIdx.x * blockDim.x + threadIdx.x;
  if (t >= Bc * Nc) return;
  int b = t / Nc, n = t % Nc;
  int m = min_idx[t];
  for (int c = 0; c < 32; ++c) {
    x2[((size_t)b * 64 + c) * Nc + n]      = f1[((size_t)b * 32 + c) * Nc + n];
    x2[((size_t)b * 64 + 32 + c) * Nc + n] = m1[((size_t)b * 32 + c) * MAc + m];
  }
}

// ---------------------------------------------------------------------------
// kNN (k=16, sorted ascending distance, ties -> lower index) node_b -> clusters
// ---------------------------------------------------------------------------
__global__ void knn_kernel(const float* __restrict__ cm,    // (B,3,MA)
                           const float* __restrict__ nb,    // (B,3,MB)
                           int* __restrict__ knn_idx) {     // (B,MB,16)
  int t = blockIdx.x * blockDim.x + threadIdx.x;
  if (t >= Bc * MBc) return;
  int b = t / MBc, mb = t % MBc;
  float bx = nb[((size_t)b * 3 + 0) * MBc + mb];
  float by = nb[((size_t)b * 3 + 1) * MBc + mb];
  float bz = nb[((size_t)b * 3 + 2) * MBc + mb];
  const float* cx = cm + ((size_t)b * 3 + 0) * MAc;
  const float* cy = cm + ((size_t)b * 3 + 1) * MAc;
  const float* cz = cm + ((size_t)b * 3 + 2) * MAc;
  unsigned long long used[4] = {0ull, 0ull, 0ull, 0ull};
  for (int k = 0; k < 16; ++k) {
    float best = 3.4e38f;
    int bi = 0;
    for (int m = 0; m < MAc; ++m) {
      if ((used[m >> 6] >> (m & 63)) & 1ull) continue;
      float dx = bx - cx[m], dy = by - cy[m], dz = bz - cz[m];
      float d = dx * dx + dy * dy + dz * dz;
      if (d < best) { best = d; bi = m; }
    }
    used[bi >> 6] |= (1ull << (bi & 63));
    knn_idx[(size_t)t * 16 + k] = bi;
  }
}

// ---------------------------------------------------------------------------
// Build knn input x = concat(pos_knn - node_b, feat_knn)  (B,67,MB*16)
// ---------------------------------------------------------------------------
__global__ void xknn_kernel(const float* __restrict__ cm,    // (B,3,MA)
                            const float* __restrict__ nb,    // (B,3,MB)
                            const float* __restrict__ naf,   // (B,64,MA)
                            const int*   __restrict__ knn_idx,
                            float* __restrict__ xk) {        // (B,67,1024)
  int t = blockIdx.x * blockDim.x + threadIdx.x;
  if (t >= Bc * MBc * 16) return;
  int b = t / (MBc * 16);
  int r = t % (MBc * 16);
  int mb = r / 16;
  int m = knn_idx[t];
  for (int c = 0; c < 3; ++c)
    xk[((size_t)b * 67 + c) * (MBc * 16) + r] =
        cm[((size_t)b * 3 + c) * MAc + m] - nb[((size_t)b * 3 + c) * MBc + mb];
  for (int ch = 0; ch < 64; ++ch)
    xk[((size_t)b * 67 + 3 + ch) * (MBc * 16) + r] =
        naf[((size_t)b * 64 + ch) * MAc + m];
}

// ---------------------------------------------------------------------------
// Build zin = concat(maxpool_k(y) broadcast, y)  (B,512,MB*16)
// ---------------------------------------------------------------------------
__global__ void zin_kernel(const float* __restrict__ y,    // (B,256,1024)
                           float* __restrict__ zin) {      // (B,512,1024)
  int t = blockIdx.x * blockDim.x + threadIdx.x;
  if (t >= Bc * 256 * MBc) return;
  int b = t / (256 * MBc);
  int r = t % (256 * MBc);
  int ch = r / MBc, mb = r % MBc;
  const float* yr = y + ((size_t)b * 256 + ch) * (MBc * 16) + mb * 16;
  float p = yr[0];
  for (int k = 1; k < 16; ++k) p = p > yr[k] ? p : yr[k];
  float* z0 = zin + ((size_t)b * 512 + ch) * (MBc * 16) + mb * 16;
  float* z1 = zin + ((size_t)b * 512 + 256 + ch) * (MBc * 16) + mb * 16;
  for (int k = 0; k < 16; ++k) { z0[k] = p; z1[k] = yr[k]; }
}

// ---------------------------------------------------------------------------
// node_b_features = maxpool_k(z); build fin input = concat(node_b, nbf)
// ---------------------------------------------------------------------------
__global__ void nbf_kernel(const float* __restrict__ z,     // (B,256,1024)
                           const float* __restrict__ nb,    // (B,3,MB)
                           float* __restrict__ nbf,         // (B,256,MB) -> d_out
                           float* __restrict__ finin) {     // (B,259,MB)
  int t = blockIdx.x * blockDim.x + threadIdx.x;
  if (t >= Bc * 256 * MBc) return;
  int b = t / (256 * MBc);
  int r = t % (256 * MBc);
  int ch = r / MBc, mb = r % MBc;
  const float* zr = z + ((size_t)b * 256 + ch) * (MBc * 16) + mb * 16;
  float p = zr[0];
  for (int k = 1; k < 16; ++k) p = p > zr[k] ? p : zr[k];
  nbf[((size_t)b * 256 + ch) * MBc + mb] = p;
  finin[((size_t)b * 259 + 3 + ch) * MBc + mb] = p;
  if (ch < 3)
    finin[((size_t)b * 259 + ch) * MBc + mb] = nb[((size_t)b * 3 + ch) * MBc + mb];
}

// ---------------------------------------------------------------------------
// global_feature = max over MB
// ---------------------------------------------------------------------------
__global__ void gmax_kernel(const float* __restrict__ fin,  // (B,1024,MB)
                            float* __restrict__ g) {        // (B,1024)
  int t = blockIdx.x * blockDim.x + threadIdx.x;
  if (t >= Bc * 1024) return;
  const float* fr = fin + (size_t)t * MBc;
  float p = fr[0];
  for (int k = 1; k < MBc; ++k) p = p > fr[k] ? p : fr[k];
  g[t] = p;
}

// ---------------------------------------------------------------------------
// Host launcher
// ---------------------------------------------------------------------------
extern "C" void kernel_launch(void* const* d_in, const int* in_sizes, int n_in,
                              void* d_out, int out_size, void* d_ws, size_t ws_size,
                              hipStream_t stream) {
  const float* pc        = (const float*)d_in[0];
  const float* intensity = (const float*)d_in[1];
  const float* sn        = (const float*)d_in[2];
  const float* label     = (const float*)d_in[3];
  const float* node_a    = (const float*)d_in[4];
  const float* node_b    = (const float*)d_in[5];
  float* out = (float*)d_out;
  float* ws  = (float*)d_ws;

  // workspace arena (floats)
  constexpr size_t P0 = 0;                       // 8,388,608
  constexpr size_t P1 = 8388608;                 // 8,388,608
  constexpr size_t P2 = 16777216;                // 2,097,152
  constexpr size_t W_SUMS  = 18874368;           // B*MA*4 = 8192
  constexpr size_t W_M1    = W_SUMS + 8192;      // B*32*MA = 65536
  constexpr size_t W_MIDX  = W_M1 + 65536;       // B*N ints = 131072
  constexpr size_t W_KIDX  = W_MIDX + 131072;    // B*MB*16 ints = 8192

  // d_out layout (tuple concatenated, float)
  constexpr size_t O_PCC = 0;                                     // (B,3,N)
  constexpr size_t O_CM  = O_PCC + (size_t)Bc * 3 * Nc;           // (B,3,MA)
  constexpr size_t O_MKI = O_CM + (size_t)Bc * 3 * MAc;           // (B,N,3)
  constexpr size_t O_F1  = O_MKI + (size_t)Bc * Nc * 3;           // (B,32,N)
  constexpr size_t O_F2  = O_F1 + (size_t)Bc * 32 * Nc;           // (B,64,N)
  constexpr size_t O_NAF = O_F2 + (size_t)Bc * 64 * Nc;           // (B,64,MA)
  constexpr size_t O_NBF = O_NAF + (size_t)Bc * 64 * MAc;         // (B,256,MB)
  constexpr size_t O_GF  = O_NBF + (size_t)Bc * 256 * MBc;        // (B,1024,1)

  int* min_idx = (int*)(ws + W_MIDX);
  int* knn_idx = (int*)(ws + W_KIDX);

  auto gemm = [&](const float* X, int wi, float* Y, int O, int I, int Ncols) {
    dim3 grid(Ncols / 64, O / 16, Bc);
    wmma_gemm_bias_relu<<<grid, 32, 0, stream>>>(
        X, (const float*)d_in[wi], (const float*)d_in[wi + 1], Y, O, I, Ncols);
  };

  // init accumulators (sums, m1, node_a_features region of d_out)
  zero_kernel<<<(8192 + 255) / 256, 256, 0, stream>>>(ws + W_SUMS, 8192);
  zero_kernel<<<(65536 + 255) / 256, 256, 0, stream>>>(ws + W_M1, 65536);
  zero_kernel<<<(131072 + 255) / 256, 256, 0, stream>>>(out + O_NAF, 131072);

  // 1) nearest-3 + segment sums
  nearest3_kernel<<<dim3(Nc / 256, Bc), 256, 0, stream>>>(
      pc, node_a, out + O_MKI, min_idx, ws + W_SUMS);
  // 2) cluster means
  cluster_mean_kernel<<<(Bc * MAc + 255) / 256, 256, 0, stream>>>(ws + W_SUMS, out + O_CM);
  // 3) pc_centers + augmented features -> P2
  pcaug_kernel<<<(Bc * Nc + 255) / 256, 256, 0, stream>>>(
      pc, intensity, sn, label, out + O_CM, min_idx, out + O_PCC, ws + P2);

  // 4) f1 MLP: 8 -> 32 -> 32 -> 32
  gemm(ws + P2, 6, ws + P0, 32, 8, Nc);
  gemm(ws + P0, 8, ws + P1, 32, 32, Nc);
  gemm(ws + P1, 10, out + O_F1, 32, 32, Nc);

  // 5) segment max of f1 -> m1; build x2 = [f1; m1[idx]] -> P0
  segmax_kernel<<<(Bc * Nc + 255) / 256, 256, 0, stream>>>(
      out + O_F1, min_idx, (unsigned*)(ws + W_M1), 32);
  x2_kernel<<<(Bc * Nc + 255) / 256, 256, 0, stream>>>(
      out + O_F1, ws + W_M1, min_idx, ws + P0);

  // 6) f2 MLP: 64 -> 64 -> 64
  gemm(ws + P0, 12, ws + P1, 64, 64, Nc);
  gemm(ws + P1, 14, out + O_F2, 64, 64, Nc);

  // 7) segment max of f2 -> node_a_features (d_out region, zero-inited)
  segmax_kernel<<<(Bc * Nc + 255) / 256, 256, 0, stream>>>(
      out + O_F2, min_idx, (unsigned*)(out + O_NAF), 64);

  // 8) kNN selection node_b -> clusters
  knn_kernel<<<(Bc * MBc + 255) / 256, 256, 0, stream>>>(out + O_CM, node_b, knn_idx);
  // 9) build knn input (B,67,1024) -> P0
  xknn_kernel<<<(Bc * MBc * 16 + 255) / 256, 256, 0, stream>>>(
      out + O_CM, node_b, out + O_NAF, knn_idx, ws + P0);

  // 10) knn1: 67 -> 256 -> 256 ; y -> P2
  gemm(ws + P0, 16, ws + P1, 256, 67, MBc * 16);
  gemm(ws + P1, 18, ws + P2, 256, 256, MBc * 16);

  // 11) zin = [maxpool(y); y] -> P0 ; knn2: 512 -> 512 -> 256 ; z -> P0
  zin_kernel<<<(Bc * 256 * MBc + 255) / 256, 256, 0, stream>>>(ws + P2, ws + P0);
  gemm(ws + P0, 20, ws + P1, 512, 512, MBc * 16);
  gemm(ws + P1, 22, ws + P0, 256, 512, MBc * 16);

  // 12) node_b_features + fin input -> P2
  nbf_kernel<<<(Bc * 256 * MBc + 255) / 256, 256, 0, stream>>>(
      ws + P0, node_b, out + O_NBF, ws + P2);

  // 13) fin: 259 -> 512 -> 1024 ; final -> P0
  gemm(ws + P2, 24, ws + P1, 512, 259, MBc);
  gemm(ws + P1, 26, ws + P0, 1024, 512, MBc);

  // 14) global max over MB
  gmax_kernel<<<(Bc * 1024 + 255) / 256, 256, 0, stream>>>(ws + P0, out + O_GF);
}